// PAAPostProcessor_22213570854934
// MI455X (gfx1250) — compile-verified
//
#include <hip/hip_runtime.h>
#include <stdint.h>

// ---------------- problem constants (match reference) ----------------
#define N_IMG   8
#define NCLS    80
#define IMH     200
#define IMW     304
#define HW      (IMH*IMW)                 // 60800
#define FLATC   (HW*NCLS)                 // 4,864,000 per image
#define TOTAL   (N_IMG*FLATC)             // 38,912,000
#define PRE_TH  0.05f
#define TOPK    1000
#define POSTK   100
#define NMS_TH  0.6f
#define IMG_W   2432.0f
#define IMG_H   1600.0f
#define LBL_OFF 4032.0f                   // IMG_W + IMG_H
#define BCLIP   4.135166556742356f        // log(1000/16)
#define NBINS   16384
#define CAP     4096                      // collect capacity per image
#define PADK    1024                      // padded NMS size

typedef __attribute__((ext_vector_type(2))) float v2f;
typedef __attribute__((ext_vector_type(8))) float v8f;

// ---------------- workspace layout (bytes) ----------------
#define OFF_HIST  ((size_t)0)                            // 8*16384*4 = 524288
#define OFF_CUT   ((size_t)524288)                       // 8*4  (pad 256)
#define OFF_CTR   (OFF_CUT + 256)                        // 8*2*4 (pad 256)
#define OFF_COL   (OFF_CTR + 256)                        // 8*4096*8 = 262144
#define OFF_BOX   (OFF_COL + (size_t)N_IMG*CAP*8)        // 8*1000*4*4 = 128000
#define OFF_SC    (OFF_BOX + (size_t)N_IMG*TOPK*4*4)     // 8*1000*4
#define OFF_LB    (OFF_SC  + (size_t)N_IMG*TOPK*4)       // 8*1000*4
#define OFF_RM    (OFF_LB  + (size_t)N_IMG*TOPK*4)       // 8*1024*32*4 = 1048576
#define OFF_KP    (OFF_RM  + (size_t)N_IMG*PADK*32*4)    // 8*32*4
#define WS_TOTAL  (OFF_KP  + (size_t)N_IMG*32*4)

__device__ __forceinline__ float sigmoidf_(float x) { return 1.0f / (1.0f + expf(-x)); }
__device__ __forceinline__ unsigned score_bin(unsigned bits) {
    unsigned b = (bits - 0x3D000000u) >> 12;    // scores > 0.05 -> bits > 0x3D4CCCCD
    return b < NBINS ? b : (NBINS - 1);
}

// ---------- pass 1: score + histogram ----------
__global__ void k_score_hist(const float* __restrict__ cls, const float* __restrict__ iou,
                             unsigned* __restrict__ hist) {
    int tid = blockIdx.x * blockDim.x + threadIdx.x;
    if (tid >= TOTAL) return;
    __builtin_prefetch(cls + tid + 65536, 0, 1);     // gfx1250 global_prefetch_b8
    int n  = tid / FLATC;
    int r  = tid - n * FLATC;
    int c  = r / HW;
    int hw = r - c * HW;
    float p = sigmoidf_(cls[tid]) * sigmoidf_(iou[n * HW + hw]);
    float s = sqrtf(p);
    if (s > PRE_TH) {
        unsigned bits = __float_as_uint(s);
        atomicAdd(&hist[n * NBINS + score_bin(bits)], 1u);
    }
}

// ---------- pass 2: per-image cutoff bin ----------
__global__ void k_cutoff(const unsigned* __restrict__ hist, int* __restrict__ cut) {
    int n = threadIdx.x;
    if (n >= N_IMG) return;
    const unsigned* h = hist + n * NBINS;
    unsigned acc = 0;
    int b;
    for (b = NBINS - 1; b > 0; --b) { acc += h[b]; if (acc >= TOPK) break; }
    cut[n] = b;
}

// ---------- pass 3: collect candidates ----------
__global__ void k_collect(const float* __restrict__ cls, const float* __restrict__ iou,
                          const int* __restrict__ cut, unsigned* __restrict__ ctr,
                          unsigned long long* __restrict__ col) {
    int tid = blockIdx.x * blockDim.x + threadIdx.x;
    if (tid >= TOTAL) return;
    int n  = tid / FLATC;
    int r  = tid - n * FLATC;
    int c  = r / HW;
    int hw = r - c * HW;
    float p = sigmoidf_(cls[tid]) * sigmoidf_(iou[n * HW + hw]);
    float s = sqrtf(p);
    if (s > PRE_TH) {
        unsigned bits = __float_as_uint(s);
        int bin = (int)score_bin(bits);
        int cb  = cut[n];
        unsigned flat = (unsigned)(hw * NCLS + c);   // permuted flatten order
        unsigned long long key =
            ((unsigned long long)bits << 24) | (unsigned long long)(0xFFFFFFu - flat);
        if (bin > cb) {
            unsigned slot = atomicAdd(&ctr[n * 2 + 0], 1u);
            if (slot < TOPK) col[(size_t)n * CAP + slot] = key;       // guaranteed < 1000
        } else if (bin == cb) {
            unsigned slot = atomicAdd(&ctr[n * 2 + 1], 1u);
            if (slot < (CAP - 1024u)) col[(size_t)n * CAP + (CAP - 1) - slot] = key;
        }
    }
}

__device__ inline void bitonic_desc(unsigned long long* k, int N, int tid, int nt) {
    for (int sz = 2; sz <= N; sz <<= 1) {
        for (int j = sz >> 1; j > 0; j >>= 1) {
            for (int i = tid; i < N; i += nt) {
                int ixj = i ^ j;
                if (ixj > i) {
                    bool desc = (i & sz) == 0;
                    unsigned long long a = k[i], b = k[ixj];
                    if (desc ? (a < b) : (a > b)) { k[i] = b; k[ixj] = a; }
                }
            }
            __syncthreads();
        }
    }
}

// ---------- pass 4: sort candidates (async->LDS), decode boxes ----------
__global__ void __launch_bounds__(1024)
k_sort_decode(const unsigned long long* __restrict__ col,
              const float* __restrict__ regs, const float* __restrict__ anch,
              float* __restrict__ boxes, float* __restrict__ scores, int* __restrict__ labels) {
    __shared__ unsigned long long keys[CAP];   // 32 KB
    int n = blockIdx.x, tid = threadIdx.x;
    // CDNA5 async global->LDS copy of candidate keys
    for (int i = tid; i < CAP; i += 1024) {
        unsigned ldsAddr = (unsigned)(uintptr_t)&keys[i];
        const void* g = (const void*)&col[(size_t)n * CAP + i];
        asm volatile("global_load_async_to_lds_b64 %0, %1, off"
                     :: "v"(ldsAddr), "v"(g) : "memory");
    }
    asm volatile("s_wait_asynccnt 0" ::: "memory");
    __syncthreads();
    bitonic_desc(keys, CAP, tid, 1024);

    if (tid < TOPK) {
        unsigned long long key = keys[tid];
        unsigned bits = (unsigned)(key >> 24);
        unsigned flat = 0xFFFFFFu - (unsigned)(key & 0xFFFFFFu);
        if (flat >= (unsigned)FLATC) flat = FLATC - 1;
        float sc = __uint_as_float(bits);
        int loc = (int)(flat / NCLS);
        int lab = (int)(flat - (unsigned)loc * NCLS) + 1;
        float a0 = anch[loc*4+0], a1 = anch[loc*4+1], a2 = anch[loc*4+2], a3 = anch[loc*4+3];
        float w = a2 - a0 + 1.0f, h = a3 - a1 + 1.0f;
        float cx = a0 + 0.5f*w, cy = a1 + 0.5f*h;
        const float* rg = regs + (size_t)n * 4 * HW;
        float dx = rg[0*HW+loc] / 10.0f, dy = rg[1*HW+loc] / 10.0f;
        float dw = fminf(rg[2*HW+loc] / 5.0f, BCLIP);
        float dh = fminf(rg[3*HW+loc] / 5.0f, BCLIP);
        float pcx = dx*w + cx, pcy = dy*h + cy;
        float pw = expf(dw)*w, ph = expf(dh)*h;
        float x1 = pcx - 0.5f*pw,        y1 = pcy - 0.5f*ph;
        float x2 = pcx + 0.5f*pw - 1.0f, y2 = pcy + 0.5f*ph - 1.0f;
        x1 = fminf(fmaxf(x1, 0.0f), IMG_W - 1.0f);
        y1 = fminf(fmaxf(y1, 0.0f), IMG_H - 1.0f);
        x2 = fminf(fmaxf(x2, 0.0f), IMG_W - 1.0f);
        y2 = fminf(fmaxf(y2, 0.0f), IMG_H - 1.0f);
        bool ok = ((x2 - x1 + 1.0f) >= 0.0f) && ((y2 - y1 + 1.0f) >= 0.0f);
        bool valid = (sc > PRE_TH) && ok;
        size_t bo = ((size_t)n * TOPK + tid) * 4;
        boxes[bo+0] = x1; boxes[bo+1] = y1; boxes[bo+2] = x2; boxes[bo+3] = y2;
        scores[(size_t)n * TOPK + tid] = valid ? sc : 0.0f;
        labels[(size_t)n * TOPK + tid] = lab;
    }
}

// ---------- pass 5: suppression matrix; WMMA computes pairwise area sums ----------
// Tile loop: tI outer (rows cached in registers, A-operand loop-invariant),
// tJ inner. Division-free IoU test: inter > TH*denom && denom > 0.
__global__ void __launch_bounds__(1024)
k_nms_rows(const float* __restrict__ boxes, const int* __restrict__ labels,
           unsigned* __restrict__ rowmask) {
    __shared__ float sx1[PADK], sy1[PADK], sx2[PADK], sy2[PADK], sar[PADK];  // 20 KB
    int n = blockIdx.x, tid = threadIdx.x;
    float x1, y1, x2, y2;
    if (tid < TOPK) {
        size_t bo = ((size_t)n * TOPK + tid) * 4;
        float off = (float)labels[(size_t)n * TOPK + tid] * LBL_OFF;
        x1 = boxes[bo+0] + off; y1 = boxes[bo+1] + off;
        x2 = boxes[bo+2] + off; y2 = boxes[bo+3] + off;
    } else {                                   // far-away degenerate pad, area 1
        x1 = y1 = x2 = y2 = -4.0e8f;
    }
    sx1[tid] = x1; sy1[tid] = y1; sx2[tid] = x2; sy2[tid] = y2;
    sar[tid] = (x2 - x1 + 1.0f) * (y2 - y1 + 1.0f);
    __syncthreads();

    int wave = tid >> 5, lane = tid & 31;
    int half = lane >> 4, nn = lane & 15;
    for (int tI = wave; tI < PADK/16; tI += 32) {        // 2 row-tiles per wave
        int iBase = tI*16 + 8*half;
        float ax1[8], ay1[8], ax2[8], ay2[8];            // row cache in VGPRs
        #pragma unroll
        for (int v = 0; v < 8; ++v) {
            ax1[v] = sx1[iBase+v]; ay1[v] = sy1[iBase+v];
            ax2[v] = sx2[iBase+v]; ay2[v] = sy2[iBase+v];
        }
        v2f a;                                           // loop-invariant A operand
        a.x = half ? 0.0f : sar[tI*16 + nn];
        a.y = half ? 0.0f : 1.0f;
        for (int tJ = 0; tJ < PADK/16; ++tJ) {
            int j = tJ*16 + nn;
            v2f b; v8f cz = {};
            b.x = half ? 0.0f : 1.0f;
            b.y = half ? 0.0f : sar[j];
            // D[m][n] = area_i[m] + area_j[n] (rank-2 outer sum)
            v8f d = __builtin_amdgcn_wmma_f32_16x16x4_f32(false, a, false, b,
                                                          (short)0, cz, false, false);
            float jx1 = sx1[j], jy1 = sy1[j], jx2 = sx2[j], jy2 = sy2[j];
            #pragma unroll
            for (int v = 0; v < 8; ++v) {
                int i = iBase + v;
                float wx = fmaxf(fminf(ax2[v], jx2) - fmaxf(ax1[v], jx1) + 1.0f, 0.0f);
                float wy = fmaxf(fminf(ay2[v], jy2) - fmaxf(ay1[v], jy1) + 1.0f, 0.0f);
                float inter = wx * wy;
                float denom = d[v] - inter;              // area_i + area_j - inter
                bool pred = (inter > NMS_TH * denom) && (denom > 0.0f) &&
                            (j > i) && (i < TOPK) && (j < TOPK);
                unsigned m = (unsigned)__ballot(pred);   // lanes 0-15: row v; 16-31: row v+8
                if (lane == 0) {
                    unsigned sh = (unsigned)((tJ & 1) * 16);
                    atomicOr(&rowmask[((size_t)n*PADK + tI*16 + v    )*32 + (tJ>>1)], (m & 0xFFFFu) << sh);
                    atomicOr(&rowmask[((size_t)n*PADK + tI*16 + v + 8)*32 + (tJ>>1)], (m >> 16)    << sh);
                }
            }
        }
    }
}

// ---------- pass 6: serial greedy scan, one wave32 per image ----------
__global__ void k_nms_serial(const unsigned* __restrict__ rowmask, unsigned* __restrict__ keepw) {
    int n = blockIdx.x, lane = threadIdx.x;    // blockDim = 32
    unsigned removed = 0u;
    for (int i = 0; i < TOPK; ++i) {
        unsigned rw = (unsigned)__shfl((int)removed, i >> 5);   // uniform scalar
        if (((rw >> (i & 31)) & 1u) == 0u)
            removed |= rowmask[((size_t)n*PADK + i)*32 + lane];
    }
    keepw[n*32 + lane] = ~removed;
}

// ---------- pass 7: final top-100 ----------
__global__ void __launch_bounds__(1024)
k_final(const float* __restrict__ boxes, const float* __restrict__ scores,
        const int* __restrict__ labels, const unsigned* __restrict__ keepw,
        float* __restrict__ out) {
    __shared__ unsigned long long keys[1024];  // 8 KB
    int n = blockIdx.x, tid = threadIdx.x;
    float s = 0.0f;
    if (tid < TOPK) {
        bool kept = (keepw[n*32 + (tid >> 5)] >> (tid & 31)) & 1u;
        s = kept ? scores[(size_t)n*TOPK + tid] : 0.0f;
    }
    keys[tid] = ((unsigned long long)__float_as_uint(s) << 32) |
                (unsigned long long)(0xFFFFFFFFu - (unsigned)tid);
    __syncthreads();
    bitonic_desc(keys, 1024, tid, 1024);
    if (tid < POSTK) {
        unsigned long long key = keys[tid];
        unsigned idx = 0xFFFFFFFFu - (unsigned)(key & 0xFFFFFFFFu);
        if (idx >= TOPK) idx = TOPK - 1;
        float sc = __uint_as_float((unsigned)(key >> 32));
        size_t bo = ((size_t)n*TOPK + idx) * 4;
        out[(size_t)n*POSTK*4 + tid*4 + 0] = boxes[bo+0];
        out[(size_t)n*POSTK*4 + tid*4 + 1] = boxes[bo+1];
        out[(size_t)n*POSTK*4 + tid*4 + 2] = boxes[bo+2];
        out[(size_t)n*POSTK*4 + tid*4 + 3] = boxes[bo+3];
        out[(size_t)N_IMG*POSTK*4 + (size_t)n*POSTK + tid] = sc;
        out[(size_t)N_IMG*POSTK*5 + (size_t)n*POSTK + tid] =
            (float)labels[(size_t)n*TOPK + idx];
    }
}

extern "C" void kernel_launch(void* const* d_in, const int* in_sizes, int n_in,
                              void* d_out, int out_size, void* d_ws, size_t ws_size,
                              hipStream_t stream) {
    const float* box_cls = (const float*)d_in[0];
    const float* box_reg = (const float*)d_in[1];
    const float* iou_prd = (const float*)d_in[2];
    const float* anchors = (const float*)d_in[3];
    char* ws = (char*)d_ws;

    unsigned*            hist = (unsigned*)(ws + OFF_HIST);
    int*                 cut  = (int*)(ws + OFF_CUT);
    unsigned*            ctr  = (unsigned*)(ws + OFF_CTR);
    unsigned long long*  col  = (unsigned long long*)(ws + OFF_COL);
    float*               bxs  = (float*)(ws + OFF_BOX);
    float*               scs  = (float*)(ws + OFF_SC);
    int*                 lbs  = (int*)(ws + OFF_LB);
    unsigned*            rmk  = (unsigned*)(ws + OFF_RM);
    unsigned*            kpw  = (unsigned*)(ws + OFF_KP);

    hipMemsetAsync(d_ws, 0, WS_TOTAL, stream);

    int blocks = (TOTAL + 255) / 256;
    k_score_hist<<<blocks, 256, 0, stream>>>(box_cls, iou_prd, hist);
    k_cutoff<<<1, 32, 0, stream>>>(hist, cut);
    k_collect<<<blocks, 256, 0, stream>>>(box_cls, iou_prd, cut, ctr, col);
    k_sort_decode<<<N_IMG, 1024, 0, stream>>>(col, box_reg, anchors, bxs, scs, lbs);
    k_nms_rows<<<N_IMG, 1024, 0, stream>>>(bxs, lbs, rmk);
    k_nms_serial<<<N_IMG, 32, 0, stream>>>(rmk, kpw);
    k_final<<<N_IMG, 1024, 0, stream>>>(bxs, scs, lbs, kpw, (float*)d_out);
}